// ChebNet_41979010351136
// MI455X (gfx1250) — compile-verified
//
#include <hip/hip_runtime.h>

#define NN 100000
#define CIN1 64
#define CH 64
#define CO 32

typedef float v2f __attribute__((ext_vector_type(2)));
typedef float v8f __attribute__((ext_vector_type(8)));

// ---------------- utility kernels ----------------

__global__ void fill_zero_kernel(float* __restrict__ p, long long n) {
    long long i = (long long)blockIdx.x * blockDim.x + threadIdx.x;
    if (i < n) p[i] = 0.0f;
}

__global__ void neg_copy_kernel(float* __restrict__ dst, const float* __restrict__ src, long long n) {
    long long i = (long long)blockIdx.x * blockDim.x + threadIdx.x;
    if (i < n) dst[i] = -src[i];
}

__global__ void degree_kernel(const long long* __restrict__ rows, float* __restrict__ deg, int E) {
    int e = blockIdx.x * blockDim.x + threadIdx.x;
    if (e < E) atomicAdd(&deg[rows[e]], 1.0f);
}

__global__ void dinv_kernel(const float* __restrict__ deg, float* __restrict__ dinv, int n) {
    int i = blockIdx.x * blockDim.x + threadIdx.x;
    if (i < n) {
        float d = deg[i];
        dinv[i] = (d > 0.0f) ? rsqrtf(fmaxf(d, 1.0f)) : 0.0f;
    }
}

__global__ void norm_kernel(const long long* __restrict__ rows, const long long* __restrict__ cols,
                            const float* __restrict__ dinv, float* __restrict__ nrm, int E) {
    int e = blockIdx.x * blockDim.x + threadIdx.x;
    if (e < E) nrm[e] = dinv[rows[e]] * dinv[cols[e]];
}

// tout[row] += -scale * norm[e] * tin[col]   (C = 64 channels, 4 per thread)
__global__ void scatter_kernel(const long long* __restrict__ rows, const long long* __restrict__ cols,
                               const float* __restrict__ nrm, const float* __restrict__ tin,
                               float* __restrict__ tout, float scale, int E) {
    int t = blockIdx.x * blockDim.x + threadIdx.x;
    int e = t >> 4;               // 16 channel-quads per edge (64 channels)
    if (e >= E) return;
    int c = (t & 15) << 2;
    float w = -scale * nrm[e];
    long long r  = rows[e];
    long long cl = cols[e];
    const float4 v = *reinterpret_cast<const float4*>(tin + cl * 64 + c);
    float* o = tout + r * 64 + c;
    atomicAdd(o + 0, w * v.x);
    atomicAdd(o + 1, w * v.y);
    atomicAdd(o + 2, w * v.z);
    atomicAdd(o + 3, w * v.w);
}

// ---------------- fused Chebyshev GEMM with f32 WMMA ----------------
// out[n, :] = relu?( T0@W[0] + T1@W[1] + T2@W[2] + bias )
// One wave32 computes one 16x16 output tile using V_WMMA_F32_16X16X4_F32.
template <int CIN, int COUT, bool RELU>
__global__ void cheb_gemm_wmma(const float* __restrict__ T0, const float* __restrict__ T1,
                               const float* __restrict__ T2, const float* __restrict__ W,
                               const float* __restrict__ bias, float* __restrict__ out,
                               int nRowTiles) {
    const int colTiles = COUT / 16;
    int wave  = blockIdx.x * (blockDim.x >> 5) + (threadIdx.x >> 5);
    int total = nRowTiles * colTiles;
    if (wave >= total) return;          // wave-uniform: EXEC stays all-ones for WMMA

    int rt = wave / colTiles;
    int ct = wave - rt * colTiles;
    int m0 = rt << 4;
    int n0 = ct << 4;

    int lane = threadIdx.x & 31;
    int half = lane >> 4;               // 0: K={0,1}, 1: K={2,3} (A); rows 0-7 / 8-15 (C/D)
    int l15  = lane & 15;
    int arow = m0 + l15;
    int bcol = n0 + l15;

    v8f acc = {};

    const float* Tk[3] = {T0, T1, T2};
    #pragma unroll
    for (int k = 0; k < 3; ++k) {
        // A: lane holds A[arow][c + 2*half + {0,1}]  (16x4 per step)
        const float* Ap = Tk[k] + (long long)arow * CIN + 2 * half;
        // B: lane holds W[k][c + 2*half + {0,1}][bcol] (4x16 per step)
        const float* Bp = W + (long long)k * CIN * COUT + (2 * half) * COUT + bcol;
        #pragma unroll
        for (int c = 0; c < CIN; c += 4) {
            v2f a = *reinterpret_cast<const v2f*>(Ap + c);   // 8B aligned: c+2*half even
            v2f b;
            b[0] = Bp[c * COUT];
            b[1] = Bp[c * COUT + COUT];
            acc = __builtin_amdgcn_wmma_f32_16x16x4_f32(
                false, a, false, b, (short)0, acc, false, false);
        }
    }

    float bv = bias[bcol];
    #pragma unroll
    for (int j = 0; j < 8; ++j) {
        int r = m0 + half * 8 + j;      // VGPR j holds rows j (lanes 0-15) and j+8 (lanes 16-31)
        float v = acc[j] + bv;
        if (RELU) v = fmaxf(v, 0.0f);
        out[(long long)r * COUT + bcol] = v;
    }
}

// ---------------- launch ----------------

extern "C" void kernel_launch(void* const* d_in, const int* in_sizes, int n_in,
                              void* d_out, int out_size, void* d_ws, size_t ws_size,
                              hipStream_t stream) {
    const float*     x    = (const float*)d_in[0];
    const long long* ei   = (const long long*)d_in[1];
    const float*     W1   = (const float*)d_in[2];
    const float*     b1   = (const float*)d_in[3];
    const float*     W2   = (const float*)d_in[4];
    const float*     b2   = (const float*)d_in[5];
    float*           outp = (float*)d_out;

    const int E = in_sizes[1] / 2;
    const long long* rows = ei;
    const long long* cols = ei + E;

    // workspace layout (floats)
    float* deg  = (float*)d_ws;              // NN
    float* dinv = deg  + NN;                 // NN
    float* nrm  = dinv + NN;                 // E
    float* buf1 = nrm  + E;                  // NN*64  (Tx1)
    float* buf2 = buf1 + (long long)NN * 64; // NN*64  (Tx2)
    float* bufH = buf2 + (long long)NN * 64; // NN*64  (hidden)

    const int TB = 256;
    const long long nfeat = (long long)NN * 64;
    const int featBlocks  = (int)((nfeat + TB - 1) / TB);
    const int edgeBlocks  = (E + TB - 1) / TB;
    const int scatBlocks  = (int)(((long long)E * 16 + TB - 1) / TB);
    const int nodeBlocks  = (NN + TB - 1) / TB;

    // ---- normalization ----
    fill_zero_kernel<<<nodeBlocks, TB, 0, stream>>>(deg, NN);
    degree_kernel<<<edgeBlocks, TB, 0, stream>>>(rows, deg, E);
    dinv_kernel<<<nodeBlocks, TB, 0, stream>>>(deg, dinv, NN);
    norm_kernel<<<edgeBlocks, TB, 0, stream>>>(rows, cols, dinv, nrm, E);

    const int rowTiles = NN / 16;  // 6250

    // ---- layer 1: Tx0 = x ----
    fill_zero_kernel<<<featBlocks, TB, 0, stream>>>(buf1, nfeat);
    scatter_kernel<<<scatBlocks, TB, 0, stream>>>(rows, cols, nrm, x, buf1, 1.0f, E);     // Tx1
    neg_copy_kernel<<<featBlocks, TB, 0, stream>>>(buf2, x, nfeat);                       // -Tx0
    scatter_kernel<<<scatBlocks, TB, 0, stream>>>(rows, cols, nrm, buf1, buf2, 2.0f, E);  // Tx2
    // h = relu(Tx0@W1[0] + Tx1@W1[1] + Tx2@W1[2] + b1) ; 6250*4 tiles / 4 waves per block
    cheb_gemm_wmma<CIN1, CH, true><<<rowTiles * (CH / 16) / 4, 128, 0, stream>>>(
        x, buf1, buf2, W1, b1, bufH, rowTiles);

    // ---- layer 2: Tx0 = h ----
    fill_zero_kernel<<<featBlocks, TB, 0, stream>>>(buf1, nfeat);
    scatter_kernel<<<scatBlocks, TB, 0, stream>>>(rows, cols, nrm, bufH, buf1, 1.0f, E);
    neg_copy_kernel<<<featBlocks, TB, 0, stream>>>(buf2, bufH, nfeat);
    scatter_kernel<<<scatBlocks, TB, 0, stream>>>(rows, cols, nrm, buf1, buf2, 2.0f, E);
    // out = Tx0@W2[0] + Tx1@W2[1] + Tx2@W2[2] + b2 ; 6250*2 tiles / 4 waves per block
    cheb_gemm_wmma<CH, CO, false><<<rowTiles * (CO / 16) / 4, 128, 0, stream>>>(
        bufH, buf1, buf2, W2, b2, outp, rowTiles);
}